// DigitCapsules_63952063037847
// MI455X (gfx1250) — compile-verified
//
#include <hip/hip_runtime.h>
#include <math.h>

// Problem constants (from reference setup_inputs)
#define BATCH  64
#define NI     4096
#define NC     32
#define ND     16      // D == d == 16
#define ICHUNK 32
#define NCHUNK (NI / ICHUNK)   // 128 i-chunks

typedef __attribute__((ext_vector_type(2))) float v2f;
typedef __attribute__((ext_vector_type(8))) float v8f;

// ---------------------------------------------------------------------------
// Fused pass: recompute u_hat via v_wmma_f32_16x16x4_f32, do routing math,
// accumulate partial s_j per (i-chunk, b-tile) block.
//   MODE 0: c_ij = 1/32 uniform (first routing iter; no logits needed)
//   MODE 1: logits = u_hat . v0
//   MODE 2: logits = u_hat . (v0 + v1)   (since b2 = uhat.v0 + uhat.v1)
// Grid: (NCHUNK, 4 b-tiles), block 256 threads = 8 waves; wave w owns
// capsules 4w..4w+3. Each WMMA C tile = one capsule's 16(b) x 16(D) slab.
// ---------------------------------------------------------------------------
template<int MODE>
__global__ __launch_bounds__(256, 1) void caps_pass(
    const float* __restrict__ x,        // [64][4096][16]
    const float* __restrict__ w,        // [4096][32][16][16]  (i,c,D,d)
    const float* __restrict__ va,       // v0  [64][32][16]  (MODE>=1)
    const float* __restrict__ vb,       // v1  [64][32][16]  (MODE==2)
    float* __restrict__ partial)        // [NCHUNK][64][32][16]
{
  const int tid   = threadIdx.x;
  const int wave  = tid >> 5;
  const int lane  = tid & 31;
  const int lhalf = lane >> 4;     // 0: lanes 0-15, 1: lanes 16-31
  const int l16   = lane & 15;
  const int ic    = blockIdx.x;    // i-chunk
  const int mt    = blockIdx.y;    // b-tile (16 batch rows)
  const int b0    = mt * 16;
  const int cbase = wave * 4;

  __shared__ float nb[16][NC];       // routing logits for this (i, b-tile)
  __shared__ float cw[16][NC + 1];   // softmax(c_ij), padded vs bank conflicts

  v8f sacc[4] = {};                  // per-wave s_j accumulators (4 capsules)

  for (int ii = 0; ii < ICHUNK; ++ii) {
    const int i = ic * ICHUNK + ii;

    // Prefetch next i's weight slab for this wave's capsules (global_prefetch)
    if (ii + 1 < ICHUNK)
      __builtin_prefetch(w + (((size_t)(i + 1) * NC + cbase) * ND) * 16, 0, 1);

    // ---- A fragments: X tile (M=16 b-rows, K=16 d). f32 16x4 frag layout:
    // v0: lanes0-15 K=4k, lanes16-31 K=4k+2 ; v1: K=4k+1 / 4k+3  -> b64/lane.
    const float* xrow = x + ((size_t)(b0 + l16) * NI + i) * ND;
    v2f afrag[4];
#pragma unroll
    for (int kk = 0; kk < 4; ++kk)
      afrag[kk] = *(const v2f*)(xrow + 4 * kk + 2 * lhalf);

    // ---- B fragments + WMMA: u_hat tiles, one per capsule this wave owns.
    // B (K=d x N=D): lanes hold N=D=l16, halves split K like A.
    v8f acc[4];
#pragma unroll
    for (int t = 0; t < 4; ++t) {
      const int c = cbase + t;
      const float* wrow = w + (((size_t)i * NC + c) * ND + l16) * 16; // row D=l16
      v2f bfrag[4];
#pragma unroll
      for (int kk = 0; kk < 4; ++kk)
        bfrag[kk] = *(const v2f*)(wrow + 4 * kk + 2 * lhalf);
      v8f a_ = {};
#pragma unroll
      for (int kk = 0; kk < 4; ++kk)
        a_ = __builtin_amdgcn_wmma_f32_16x16x4_f32(
            false, afrag[kk], false, bfrag[kk], (short)0, a_, false, false);
      acc[t] = a_;
    }

    if (MODE > 0) {
      // ---- agreement logits: a[b,c] = sum_D u_hat * vsum, via lane-half
      // butterfly (masks 1/2/4/8 never cross the 16-lane halves).
#pragma unroll
      for (int t = 0; t < 4; ++t) {
        const int c = cbase + t;
#pragma unroll
        for (int r = 0; r < 8; ++r) {
          const int b = b0 + r + 8 * lhalf;       // C-matrix row for this half
          const size_t vi = ((size_t)b * NC + c) * ND + l16;
          float vv = va[vi];
          if (MODE == 2) vv += vb[vi];
          float p = acc[t][r] * vv;
          p += __shfl_xor(p, 1, 32);
          p += __shfl_xor(p, 2, 32);
          p += __shfl_xor(p, 4, 32);
          p += __shfl_xor(p, 8, 32);
          if (l16 == 0) nb[r + 8 * lhalf][c] = p;
        }
      }
      __syncthreads();
      // ---- softmax over the 32 capsules, one thread per batch row
      if (tid < 16) {
        float m = -3.4e38f;
#pragma unroll
        for (int c = 0; c < NC; ++c) m = fmaxf(m, nb[tid][c]);
        float s = 0.0f;
#pragma unroll
        for (int c = 0; c < NC; ++c) s += __expf(nb[tid][c] - m);
        const float inv = 1.0f / s;
#pragma unroll
        for (int c = 0; c < NC; ++c) cw[tid][c] = __expf(nb[tid][c] - m) * inv;
      }
      __syncthreads();
    }

    // ---- s_j accumulation: sacc += c_ij * u_hat (per-lane, tile layout)
#pragma unroll
    for (int t = 0; t < 4; ++t) {
#pragma unroll
      for (int r = 0; r < 8; ++r) {
        const float cweight =
            (MODE == 0) ? (1.0f / 32.0f) : cw[r + 8 * lhalf][cbase + t];
        sacc[t][r] += cweight * acc[t][r];
      }
    }
  }

  // ---- write per-block partial s_j (disjoint, fully overwritten -> no init)
#pragma unroll
  for (int t = 0; t < 4; ++t) {
    const int c = cbase + t;
#pragma unroll
    for (int r = 0; r < 8; ++r) {
      const int b = b0 + r + 8 * lhalf;
      partial[(((size_t)ic * BATCH + b) * NC + c) * ND + l16] = sacc[t][r];
    }
  }
}

// ---------------------------------------------------------------------------
// Deterministic reduction over i-chunks + squash. One thread per (b,c) row.
// ---------------------------------------------------------------------------
__global__ __launch_bounds__(256) void caps_reduce(
    const float* __restrict__ partial,  // [NCHUNK][64][32][16]
    float* __restrict__ vout)           // [64][32][16]
{
  const int row = blockIdx.x * blockDim.x + threadIdx.x;  // b*32+c
  if (row >= BATCH * NC) return;
  float s[ND];
#pragma unroll
  for (int d = 0; d < ND; ++d) s[d] = 0.0f;
  for (int ch = 0; ch < NCHUNK; ++ch) {
    const float* p = partial + ((size_t)ch * BATCH * NC + row) * ND;
#pragma unroll
    for (int d = 0; d < ND; ++d) s[d] += p[d];
  }
  float sq = 0.0f;
#pragma unroll
  for (int d = 0; d < ND; ++d) sq += s[d] * s[d];
  const float scale = 1.0f / ((1.0f + sq) * sqrtf(sq + 1e-9f));
#pragma unroll
  for (int d = 0; d < ND; ++d) vout[(size_t)row * ND + d] = s[d] * scale;
}

// ---------------------------------------------------------------------------
extern "C" void kernel_launch(void* const* d_in, const int* in_sizes, int n_in,
                              void* d_out, int out_size, void* d_ws, size_t ws_size,
                              hipStream_t stream) {
  (void)in_sizes; (void)n_in; (void)out_size; (void)ws_size;
  const float* x = (const float*)d_in[0];   // [64,4096,16]
  const float* w = (const float*)d_in[1];   // [1,4096,32,16,16]
  float* out = (float*)d_out;               // [64,32,16]

  // workspace: partials (NCHUNK*64*32*16 floats = 16.8MB) + v0 + v1
  float* P  = (float*)d_ws;
  float* v0 = P + (size_t)NCHUNK * BATCH * NC * ND;
  float* v1 = v0 + (size_t)BATCH * NC * ND;

  const dim3 grid(NCHUNK, 4), blk(256);
  const dim3 rgrid((BATCH * NC + 255) / 256), rblk(256);

  caps_pass<0><<<grid, blk, 0, stream>>>(x, w, nullptr, nullptr, P);
  caps_reduce<<<rgrid, rblk, 0, stream>>>(P, v0);
  caps_pass<1><<<grid, blk, 0, stream>>>(x, w, v0, nullptr, P);
  caps_reduce<<<rgrid, rblk, 0, stream>>>(P, v1);
  caps_pass<2><<<grid, blk, 0, stream>>>(x, w, v0, v1, P);
  caps_reduce<<<rgrid, rblk, 0, stream>>>(P, out);
}